// GCN_25202868093367
// MI455X (gfx1250) — compile-verified
//
#include <hip/hip_runtime.h>
#include <math.h>

// ---------------------------------------------------------------------------
// GCN forward on MI455X (gfx1250, wave32).
//   h1   = x @ W1                    (WMMA f32 16x16x4, W1 staged via TDM)
//   a1   = scatter_add(h1[s]*norm)   (wave-per-edge, f32 atomics)
//   h1a  = tanh(a1 + h1*dis^2 + b1)  (self-loop folded in)
//   h2   = h1a @ W2  (128x2)
//   a2   = scatter_add(h2[s]*norm)
//   emb  = tanh(a2 + h2*dis^2 + b2)
//   out  = sigmoid(emb @ Wc + bc)
// ---------------------------------------------------------------------------

typedef __attribute__((ext_vector_type(2))) float v2f;
typedef __attribute__((ext_vector_type(8))) float v8f;
typedef __attribute__((ext_vector_type(4))) unsigned int v4u;
typedef __attribute__((ext_vector_type(8))) int v8i;
typedef __attribute__((ext_vector_type(4))) int v4i;

#define FDIM 128

// ---------------- degree / norm ----------------

__global__ __launch_bounds__(256) void k_init_deg(float* deg, int n) {
    int v = blockIdx.x * 256 + threadIdx.x;
    if (v < n) deg[v] = 1.0f;  // self-loop contribution
}

__global__ __launch_bounds__(256) void k_deg(const int* __restrict__ dst, int E,
                                             float* __restrict__ deg) {
    int e = blockIdx.x * 256 + threadIdx.x;
    if (e < E) atomicAdd(&deg[dst[e]], 1.0f);
}

__global__ __launch_bounds__(256) void k_dis(float* deg, int n) {
    int v = blockIdx.x * 256 + threadIdx.x;
    if (v < n) {
        float d = deg[v];
        deg[v] = (d > 0.0f) ? rsqrtf(d) : 0.0f;
    }
}

__global__ __launch_bounds__(256) void k_norm(const int* __restrict__ src,
                                              const int* __restrict__ dst,
                                              const float* __restrict__ dis,
                                              float* __restrict__ norm, int E) {
    int e = blockIdx.x * 256 + threadIdx.x;
    if (e < E) norm[e] = dis[src[e]] * dis[dst[e]];
}

// ---------------- GEMM1: h1 = x @ W1 via V_WMMA_F32_16X16X4_F32 ----------------
// Block: 256 threads = 8 waves. Each wave computes a 16x128 strip of h1.
// W1 (128x128 fp32, 64KB) DMA'd into LDS once per block by the Tensor Data
// Mover (tensor_load_to_lds, tracked with TENSORcnt).

__global__ __launch_bounds__(256) void k_gemm1_wmma(const float* __restrict__ x,
                                                    const float* __restrict__ W,
                                                    float* __restrict__ h,
                                                    int n) {
    __shared__ float sW[FDIM * FDIM];  // 64 KB
    int tid = threadIdx.x;

#if __has_builtin(__builtin_amdgcn_tensor_load_to_lds)
    if (tid < 32) {
        // Tensor DMA descriptor (ISA cdna5 §8): 2-D 128x128 tile of 4-byte data.
        unsigned lds_off = (unsigned)(uintptr_t)(void*)sW;   // flat LDS addr: low 32b = LDS offset
        unsigned long long ga = (unsigned long long)(uintptr_t)W;
        v4u g0;
        g0[0] = 1u;                                          // count=1 (valid user D#)
        g0[1] = lds_off;                                     // lds_addr (bytes)
        g0[2] = (unsigned)(ga & 0xFFFFFFFFu);                // global_addr[31:0]
        g0[3] = (unsigned)((ga >> 32) & 0x1FFFFFFu)          // global_addr[56:32]
                | (2u << 30);                                // type=2 ("image")
        v8i g1;
        g1[0] = 0x00020000;                                  // data_size=2 (4 bytes)
        g1[1] = (int)(128u << 16);                           // tensor_dim0 = 128 (lo16)
        g1[2] = (int)(128u << 16);                           // dim0 hi=0 | tensor_dim1 lo16 = 128
        g1[3] = (int)(128u << 16);                           // dim1 hi=0 | tile_dim0 = 128
        g1[4] = 128;                                         // tile_dim1 = 128, tile_dim2 = 0
        g1[5] = 128;                                         // tensor_dim0_stride = 128 (lo32)
        g1[6] = 0;
        g1[7] = 0;
        v4i z4 = {};
        v8i z8 = {};
        __builtin_amdgcn_tensor_load_to_lds(g0, g1, z4, z4, z8, 0);
        __builtin_amdgcn_s_wait_tensorcnt(0);
    }
#else
    for (int i = tid * 4; i < FDIM * FDIM; i += 256 * 4) {
        *(float4*)&sW[i] = *(const float4*)&W[i];
    }
#endif
    __syncthreads();

    int wave  = tid >> 5;
    int lane  = tid & 31;
    int lhalf = lane >> 4;       // 0: K, K+1   1: K+2, K+3 (32-bit A layout)
    int m     = lane & 15;

    int row = blockIdx.x * 128 + wave * 16 + m;
    float valid = (row < n) ? 1.0f : 0.0f;       // branchless tail: EXEC stays all-1s
    int rowc = (row < n) ? row : (n - 1);
    const float* xrow = x + (size_t)rowc * FDIM;

    v8f acc[8] = {};

    for (int k0 = 0; k0 < FDIM; k0 += 4) {
        int ka = k0 + (lhalf << 1);
        v2f a;
        a.x = xrow[ka] * valid;
        a.y = xrow[ka + 1] * valid;
#pragma unroll
        for (int t = 0; t < 8; ++t) {
            int col = t * 16 + m;
            v2f b;
            b.x = sW[ka * FDIM + col];
            b.y = sW[(ka + 1) * FDIM + col];
            acc[t] = __builtin_amdgcn_wmma_f32_16x16x4_f32(
                false, a, false, b, (short)0, acc[t], false, false);
        }
    }

    // C/D layout: VGPR r -> lanes 0-15: M=r, lanes 16-31: M=8+r; N = lane&15
    int mrow0 = blockIdx.x * 128 + wave * 16 + lhalf * 8;
#pragma unroll
    for (int r = 0; r < 8; ++r) {
        int rr = mrow0 + r;
        if (rr < n) {
            float* hrow = h + (size_t)rr * FDIM + m;
#pragma unroll
            for (int t = 0; t < 8; ++t) hrow[t * 16] = acc[t][r];
        }
    }
}

// ---------------- layer-1 scatter: one wave (32 lanes) per edge ----------------

__global__ __launch_bounds__(256) void k_scatter1(const int* __restrict__ src,
                                                  const int* __restrict__ dst,
                                                  const float* __restrict__ norm,
                                                  const float* __restrict__ h1,
                                                  float* __restrict__ agg, int E) {
    long long idx = (long long)blockIdx.x * 256 + threadIdx.x;
    int e = (int)(idx >> 5);
    if (e >= E) return;
    int c0 = ((int)idx & 31) * 4;
    int s = src[e], d = dst[e];
    float w = norm[e];
    float4 hv = *(const float4*)(h1 + (size_t)s * FDIM + c0);
    float* out = agg + (size_t)d * FDIM + c0;
    atomicAdd(out + 0, hv.x * w);
    atomicAdd(out + 1, hv.y * w);
    atomicAdd(out + 2, hv.z * w);
    atomicAdd(out + 3, hv.w * w);
}

// h1a = tanh(agg + h1*dis^2 + b1), written back into agg in place.
__global__ __launch_bounds__(256) void k_fin1(float* __restrict__ agg,
                                              const float* __restrict__ h1,
                                              const float* __restrict__ dis,
                                              const float* __restrict__ b1,
                                              int n) {
    long long idx = (long long)blockIdx.x * 256 + threadIdx.x;
    if (idx >= (long long)n * FDIM) return;
    int v = (int)(idx >> 7);
    int c = (int)(idx & 127);
    float dv = dis[v];
    agg[idx] = tanhf(agg[idx] + h1[idx] * dv * dv + b1[c]);
}

// ---------------- layer 2: h2 = h1a @ W2 (128x2) ----------------

__global__ __launch_bounds__(256) void k_gemm2(const float* __restrict__ h1a,
                                               const float* __restrict__ W2,
                                               float* __restrict__ h2, int n) {
    __shared__ float sW[FDIM * 2];
    if (threadIdx.x < FDIM * 2) sW[threadIdx.x] = W2[threadIdx.x];
    __syncthreads();
    int v = blockIdx.x * 256 + threadIdx.x;
    if (v >= n) return;
    const float* row = h1a + (size_t)v * FDIM;
    float a0 = 0.0f, a1 = 0.0f;
#pragma unroll 4
    for (int k = 0; k < FDIM; k += 4) {
        float4 xv = *(const float4*)(row + k);
        a0 += xv.x * sW[2 * k]     + xv.y * sW[2 * k + 2]
            + xv.z * sW[2 * k + 4] + xv.w * sW[2 * k + 6];
        a1 += xv.x * sW[2 * k + 1] + xv.y * sW[2 * k + 3]
            + xv.z * sW[2 * k + 5] + xv.w * sW[2 * k + 7];
    }
    h2[2 * v] = a0;
    h2[2 * v + 1] = a1;
}

__global__ __launch_bounds__(256) void k_scatter2(const int* __restrict__ src,
                                                  const int* __restrict__ dst,
                                                  const float* __restrict__ norm,
                                                  const float* __restrict__ h2,
                                                  float* __restrict__ agg2, int E) {
    int e = blockIdx.x * 256 + threadIdx.x;
    if (e >= E) return;
    int s = src[e], d = dst[e];
    float w = norm[e];
    atomicAdd(&agg2[2 * d], h2[2 * s] * w);
    atomicAdd(&agg2[2 * d + 1], h2[2 * s + 1] * w);
}

__global__ __launch_bounds__(256) void k_fin2(const float* __restrict__ agg2,
                                              const float* __restrict__ h2,
                                              const float* __restrict__ dis,
                                              const float* __restrict__ b2,
                                              const float* __restrict__ Wc,
                                              const float* __restrict__ bc,
                                              float* __restrict__ out, int n) {
    int v = blockIdx.x * 256 + threadIdx.x;
    if (v >= n) return;
    float dv = dis[v];
    float d2 = dv * dv;
    float e0 = tanhf(agg2[2 * v] + h2[2 * v] * d2 + b2[0]);
    float e1 = tanhf(agg2[2 * v + 1] + h2[2 * v + 1] * d2 + b2[1]);
    float z = e0 * Wc[0] + e1 * Wc[1] + bc[0];
    out[v] = 1.0f / (1.0f + expf(-z));
}

// ---------------- launcher ----------------

extern "C" void kernel_launch(void* const* d_in, const int* in_sizes, int n_in,
                              void* d_out, int out_size, void* d_ws, size_t ws_size,
                              hipStream_t stream) {
    const float* x  = (const float*)d_in[0];
    const int*   ei = (const int*)d_in[1];
    const float* W1 = (const float*)d_in[2];
    const float* b1 = (const float*)d_in[3];
    const float* W2 = (const float*)d_in[4];
    const float* b2 = (const float*)d_in[5];
    const float* Wc = (const float*)d_in[6];
    const float* bc = (const float*)d_in[7];

    const int N = in_sizes[0] / FDIM;
    const int E = in_sizes[1] / 2;
    const int* src = ei;
    const int* dst = ei + E;

    // workspace carve-out
    char* ws = (char*)d_ws;
    size_t off = 0;
    auto carve = [&](size_t bytes) -> void* {
        void* p = ws + off;
        off += (bytes + 255) & ~(size_t)255;
        return p;
    };
    float* dis  = (float*)carve((size_t)N * 4);            // deg -> dis in place
    float* norm = (float*)carve((size_t)E * 4);
    float* h1   = (float*)carve((size_t)N * FDIM * 4);
    float* agg1 = (float*)carve((size_t)N * FDIM * 4);     // becomes h1a in place
    float* h2   = (float*)carve((size_t)N * 2 * 4);
    float* agg2 = (float*)carve((size_t)N * 2 * 4);

    (void)hipMemsetAsync(agg1, 0, (size_t)N * FDIM * 4, stream);
    (void)hipMemsetAsync(agg2, 0, (size_t)N * 2 * 4, stream);

    int gN = (N + 255) / 256;
    int gE = (E + 255) / 256;

    k_init_deg<<<gN, 256, 0, stream>>>(dis, N);
    k_deg<<<gE, 256, 0, stream>>>(dst, E, dis);
    k_dis<<<gN, 256, 0, stream>>>(dis, N);
    k_norm<<<gE, 256, 0, stream>>>(src, dst, dis, norm, E);

    k_gemm1_wmma<<<(N + 127) / 128, 256, 0, stream>>>(x, W1, h1, N);

    long long s1_threads = (long long)E * 32;
    k_scatter1<<<(unsigned)((s1_threads + 255) / 256), 256, 0, stream>>>(
        src, dst, norm, h1, agg1, E);

    long long f1_threads = (long long)N * FDIM;
    k_fin1<<<(unsigned)((f1_threads + 255) / 256), 256, 0, stream>>>(
        agg1, h1, dis, b1, N);

    k_gemm2<<<gN, 256, 0, stream>>>(agg1, W2, h2, N);
    k_scatter2<<<gE, 256, 0, stream>>>(src, dst, norm, h2, agg2, E);
    k_fin2<<<gN, 256, 0, stream>>>(agg2, h2, dis, b2, Wc, bc, (float*)d_out, N);
}